// NTM_77318001263252
// MI455X (gfx1250) — compile-verified
//
#include <hip/hip_runtime.h>
#include <math.h>

#define DEVFN __device__ __forceinline__

namespace {

constexpr int   B_    = 32;
constexpr int   S_    = 16384;
constexpr int   M_    = 64;
constexpr int   HID_  = 512;
constexpr int   KL_   = 192;    // LSTM_IN = 64+64+64
constexpr int   NZ_   = 2048;   // 4*HID
constexpr int   NR_   = 70;     // N_READ*(M+6)
constexpr int   NRP_  = 80;     // padded to 16
constexpr int   NW_   = 198;    // 3*M+6
constexpr int   NWP_  = 208;    // padded to 16
constexpr int   EXT_  = 576;    // HID + M
constexpr int   RT_   = 1024;   // threads for the softmax-chain reductions
constexpr float EPS_  = 1e-8f;

typedef float v2f __attribute__((ext_vector_type(2)));
typedef float v8f __attribute__((ext_vector_type(8)));

DEVFN float sigm(float x)  { return 1.0f / (1.0f + expf(-x)); }
DEVFN float softp(float x) { return (x > 20.0f) ? x : log1pf(expf(x)); }

// ---------------------------------------------------------------------------
// K1: xt = concat(x_input, prev_desired_output, prev_read_vec)  (B,192)
// ---------------------------------------------------------------------------
__global__ void build_xt_kernel(const float* __restrict__ x,
                                const float* __restrict__ pdo,
                                const float* __restrict__ prv,
                                float* __restrict__ xt) {
  int t = blockIdx.x * blockDim.x + threadIdx.x;
  if (t >= B_ * KL_) return;
  int b = t / KL_, j = t % KL_;
  float v;
  if (j < 64)       v = x  [b * 64 + j];
  else if (j < 128) v = pdo[b * 64 + (j - 64)];
  else              v = prv[b * 64 + (j - 128)];
  xt[t] = v;
}

// ---------------------------------------------------------------------------
// K2/K4: WMMA GEMM:  C[m,n] = sum_k A[m,k] * W[n,k] + bias[n] (+bias2[n])
// A: (32 x K) row-major, W: (N x K) row-major. One wave per 16x16 tile.
// V_WMMA_F32_16X16X4_F32; ISA layouts:
//   A 16x4 : lane M = lane&15, K pair = 2*(lane>>4)          (2 VGPRs)
//   B 4x16 : lane N = lane&15, K pair = 2*(lane>>4)          (2 VGPRs)
//   C/D    : row M = vgpr + 8*(lane>>4), col N = lane&15     (8 VGPRs)
// Full tiles (uniform branch) take an unmasked hot loop: load->load->wmma.
// ---------------------------------------------------------------------------
__global__ void wmma_gemm_nt(const float* __restrict__ A,
                             const float* __restrict__ W,
                             const float* __restrict__ bias,
                             const float* __restrict__ bias2,
                             float* __restrict__ C,
                             int K, int N, int ldC) {
  const int lane = threadIdx.x;          // 0..31
  const int sub  = lane & 15;            // M index (for A) / N index (for B)
  const int half = lane >> 4;            // K pair select: 0 -> K=0,1 ; 1 -> K=2,3
  const int n0   = blockIdx.x * 16;
  const int m0   = blockIdx.y * 16;
  const int n    = n0 + sub;

  const float* Arow = A + (size_t)(m0 + sub) * K + 2 * half;

  v8f acc;

  if (n0 + 16 <= N) {
    // ---- full tile: no masking anywhere ----
    float bn = bias[n];
    if (bias2) bn += bias2[n];
#pragma unroll
    for (int i = 0; i < 8; ++i) acc[i] = bn;

    const float* Wrow = W + (size_t)n * K + 2 * half;
    for (int k0 = 0; k0 < K; k0 += 4) {
      float2 a2 = *reinterpret_cast<const float2*>(Arow + k0);
      float2 b2 = *reinterpret_cast<const float2*>(Wrow + k0);
      v2f av; av[0] = a2.x; av[1] = a2.y;
      v2f bv; bv[0] = b2.x; bv[1] = b2.y;
      acc = __builtin_amdgcn_wmma_f32_16x16x4_f32(
          false, av, false, bv, (short)0, acc, false, false);
    }
#pragma unroll
    for (int v = 0; v < 8; ++v)
      C[(size_t)(m0 + v + 8 * half) * ldC + n] = acc[v];
  } else {
    // ---- ragged tail tile: clamp loads, mask padded columns ----
    const int   nc = (n < N) ? n : (N - 1);
    const float nm = (n < N) ? 1.0f : 0.0f;
    float bn = (n < N) ? bias[n] : 0.0f;
    if (bias2 && n < N) bn += bias2[n];
#pragma unroll
    for (int i = 0; i < 8; ++i) acc[i] = bn;

    const float* Wrow = W + (size_t)nc * K + 2 * half;
    for (int k0 = 0; k0 < K; k0 += 4) {
      float2 a2 = *reinterpret_cast<const float2*>(Arow + k0);
      float2 b2 = *reinterpret_cast<const float2*>(Wrow + k0);
      v2f av; av[0] = a2.x;      av[1] = a2.y;
      v2f bv; bv[0] = b2.x * nm; bv[1] = b2.y * nm;
      acc = __builtin_amdgcn_wmma_f32_16x16x4_f32(
          false, av, false, bv, (short)0, acc, false, false);
    }
    if (n < N) {
#pragma unroll
      for (int v = 0; v < 8; ++v)
        C[(size_t)(m0 + v + 8 * half) * ldC + n] = acc[v];
    }
  }
}

// ---------------------------------------------------------------------------
// K3: LSTM activations. h = sigmoid(o) * tanh(sigmoid(i) * tanh(g))
// ---------------------------------------------------------------------------
__global__ void lstm_act_kernel(const float* __restrict__ z,
                                float* __restrict__ h) {
  int t = blockIdx.x * blockDim.x + threadIdx.x;
  if (t >= B_ * HID_) return;
  int b = t / HID_, j = t % HID_;
  const float* zb = z + (size_t)b * NZ_;
  float c = sigm(zb[j]) * tanhf(zb[1024 + j]);
  h[t] = sigm(zb[1536 + j]) * tanhf(c);
}

// ---------------------------------------------------------------------------
// K5: per-batch addressing parameters.
// prm[(head*B+b)*8] = {beta, g, s0, s1, s2, gamma, kn, 0}
// ---------------------------------------------------------------------------
__global__ void params_kernel(const float* __restrict__ outr,
                              const float* __restrict__ outw,
                              float* __restrict__ kr, float* __restrict__ kw,
                              float* __restrict__ er, float* __restrict__ ad,
                              float* __restrict__ prm) {
  int b = threadIdx.x;
  if (b >= B_) return;

  // read head
  {
    const float* o = outr + (size_t)b * NRP_;
    float s = 0.0f;
    for (int m = 0; m < M_; ++m) { float v = o[m]; kr[b * M_ + m] = v; s += v * v; }
    float* p = prm + (size_t)b * 8;
    p[0] = softp(o[64]);
    p[1] = sigm(o[65]);
    float a0 = softp(o[66]), a1 = softp(o[67]), a2 = softp(o[68]);
    float mx = fmaxf(a0, fmaxf(a1, a2));
    float e0 = expf(a0 - mx), e1 = expf(a1 - mx), e2 = expf(a2 - mx);
    float den = e0 + e1 + e2;
    p[2] = e0 / den; p[3] = e1 / den; p[4] = e2 / den;
    p[5] = 1.0f + softp(o[69]);
    p[6] = fmaxf(sqrtf(s), EPS_);
    p[7] = 0.0f;
  }
  // write head
  {
    const float* o = outw + (size_t)b * NWP_;
    float s = 0.0f;
    for (int m = 0; m < M_; ++m) { float v = o[m]; kw[b * M_ + m] = v; s += v * v; }
    float* p = prm + (size_t)(B_ + b) * 8;
    p[0] = softp(o[64]);
    p[1] = sigm(o[65]);
    float a0 = softp(o[66]), a1 = softp(o[67]), a2 = softp(o[68]);
    float mx = fmaxf(a0, fmaxf(a1, a2));
    float e0 = expf(a0 - mx), e1 = expf(a1 - mx), e2 = expf(a2 - mx);
    float den = e0 + e1 + e2;
    p[2] = e0 / den; p[3] = e1 / den; p[4] = e2 / den;
    p[5] = 1.0f + softp(o[69]);
    p[6] = fmaxf(sqrtf(s), EPS_);
    p[7] = 0.0f;
    for (int m = 0; m < M_; ++m) {
      er[b * M_ + m] = o[70 + m];
      ad[b * M_ + m] = o[134 + m];
    }
  }
}

// ---------------------------------------------------------------------------
// K6: streaming pass 1 over memory (128 MB, coalesced 512B per wave-iter):
//   z_r[b,s] = beta_r * (k_r . mem[b,s]) / (kn_r * max(||mem[b,s]||, EPS))
//   z_w likewise. 16 lanes per row, 2 rows per wave.
// ---------------------------------------------------------------------------
__global__ void scores_kernel(const float* __restrict__ mem,
                              const float* __restrict__ kr,
                              const float* __restrict__ kw,
                              const float* __restrict__ prm,
                              float* __restrict__ zr,
                              float* __restrict__ zw) {
  const int b     = blockIdx.x >> 5;
  const int chunk = blockIdx.x & 31;      // 512 rows per block
  const int wave  = threadIdx.x >> 5;
  const int lane  = threadIdx.x & 31;
  const int sub   = lane & 15;
  const int half  = lane >> 4;

  const float4 k4r = *reinterpret_cast<const float4*>(kr + b * M_ + sub * 4);
  const float4 k4w = *reinterpret_cast<const float4*>(kw + b * M_ + sub * 4);
  const float cR = prm[(size_t)b * 8 + 0]        / prm[(size_t)b * 8 + 6];
  const float cW = prm[(size_t)(B_ + b) * 8 + 0] / prm[(size_t)(B_ + b) * 8 + 6];

  const float* memb = mem + (size_t)b * S_ * M_;

  for (int it = 0; it < 32; ++it) {
    int s = chunk * 512 + it * 16 + wave * 2 + half;
    const float4 v = *reinterpret_cast<const float4*>(memb + (size_t)s * M_ + sub * 4);
    float pr = v.x * k4r.x + v.y * k4r.y + v.z * k4r.z + v.w * k4r.w;
    float pw = v.x * k4w.x + v.y * k4w.y + v.z * k4w.z + v.w * k4w.w;
    float sq = v.x * v.x + v.y * v.y + v.z * v.z + v.w * v.w;
#pragma unroll
    for (int off = 1; off < 16; off <<= 1) {
      pr += __shfl_xor(pr, off, 32);
      pw += __shfl_xor(pw, off, 32);
      sq += __shfl_xor(sq, off, 32);
    }
    if (sub == 0) {
      float inv = 1.0f / fmaxf(sqrtf(sq), EPS_);
      zr[(size_t)b * S_ + s] = cR * pr * inv;
      zw[(size_t)b * S_ + s] = cW * pw * inv;
    }
  }
}

// ---------------------------------------------------------------------------
// K7: per-(head,b) max over S (one block -> deterministic)
// ---------------------------------------------------------------------------
__global__ void maxred_kernel(const float* __restrict__ zr,
                              const float* __restrict__ zw,
                              float* __restrict__ zmax) {
  int head = blockIdx.x / B_, b = blockIdx.x % B_;
  const float* z = (head ? zw : zr) + (size_t)b * S_;
  float m = -INFINITY;
  for (int i = threadIdx.x; i < S_; i += RT_) m = fmaxf(m, z[i]);
  __shared__ float red[RT_];
  red[threadIdx.x] = m;
  __syncthreads();
  for (int off = RT_ / 2; off > 0; off >>= 1) {
    if ((int)threadIdx.x < off)
      red[threadIdx.x] = fmaxf(red[threadIdx.x], red[threadIdx.x + off]);
    __syncthreads();
  }
  if (threadIdx.x == 0) zmax[blockIdx.x] = red[0];
}

// ---------------------------------------------------------------------------
// K8: in-place exp(z - zmax), per-(head,b) sum (one block -> deterministic)
// ---------------------------------------------------------------------------
__global__ void expsum_kernel(float* __restrict__ zr, float* __restrict__ zw,
                              const float* __restrict__ zmax,
                              float* __restrict__ esum) {
  int head = blockIdx.x / B_, b = blockIdx.x % B_;
  float* z = (head ? zw : zr) + (size_t)b * S_;
  const float mx = zmax[blockIdx.x];
  float s = 0.0f;
  for (int i = threadIdx.x; i < S_; i += RT_) {
    float e = expf(z[i] - mx);
    z[i] = e;
    s += e;
  }
  __shared__ float red[RT_];
  red[threadIdx.x] = s;
  __syncthreads();
  for (int off = RT_ / 2; off > 0; off >>= 1) {
    if ((int)threadIdx.x < off) red[threadIdx.x] += red[threadIdx.x + off];
    __syncthreads();
  }
  if (threadIdx.x == 0) esum[blockIdx.x] = red[0];
}

// ---------------------------------------------------------------------------
// K9: circular shift + sharpen: wgam = (g/esum * (s0*e[s-1]+s1*e[s]+s2*e[s+1]))^gamma
// ---------------------------------------------------------------------------
__global__ void shiftpow_kernel(const float* __restrict__ zr,
                                const float* __restrict__ zw,
                                const float* __restrict__ prm,
                                const float* __restrict__ esum,
                                float* __restrict__ wgr, float* __restrict__ wgw,
                                float* __restrict__ wsum) {
  int head = blockIdx.x / B_, b = blockIdx.x % B_;
  const float* p = prm + (size_t)(head * B_ + b) * 8;
  const float g = p[1], s0 = p[2], s1 = p[3], s2 = p[4], gamma = p[5];
  const float cg = g / esum[blockIdx.x];
  const float* e = (head ? zw : zr) + (size_t)b * S_;
  float* wg      = (head ? wgw : wgr) + (size_t)b * S_;

  float sum = 0.0f;
  for (int i = threadIdx.x; i < S_; i += RT_) {
    int im1 = (i == 0) ? (S_ - 1) : (i - 1);
    int ip1 = (i == S_ - 1) ? 0 : (i + 1);
    float wsv  = cg * (s0 * e[im1] + s1 * e[i] + s2 * e[ip1]);
    float wgam = powf(wsv, gamma);
    wg[i] = wgam;
    sum += wgam;
  }
  __shared__ float red[RT_];
  red[threadIdx.x] = sum;
  __syncthreads();
  for (int off = RT_ / 2; off > 0; off >>= 1) {
    if ((int)threadIdx.x < off) red[threadIdx.x] += red[threadIdx.x + off];
    __syncthreads();
  }
  if (threadIdx.x == 0) wsum[blockIdx.x] = red[0];
}

// ---------------------------------------------------------------------------
// K10: streaming pass 2 (memory re-read hits L2):
//   new_mem[b,s,:] = mem*(1 - wW*erase) + wW*add     (written to d_out)
//   r partials per (b,chunk) to workspace (deterministic, no atomics)
// ---------------------------------------------------------------------------
__global__ void finalize_kernel(const float* __restrict__ mem,
                                const float* __restrict__ wgr,
                                const float* __restrict__ wgw,
                                const float* __restrict__ wsum,
                                const float* __restrict__ er,
                                const float* __restrict__ ad,
                                float* __restrict__ newmem,
                                float* __restrict__ rpart) {
  const int b     = blockIdx.x >> 5;
  const int chunk = blockIdx.x & 31;
  const int wave  = threadIdx.x >> 5;
  const int lane  = threadIdx.x & 31;
  const int sub   = lane & 15;
  const int half  = lane >> 4;

  const float invR = 1.0f / wsum[b];
  const float invW = 1.0f / wsum[B_ + b];
  const float4 e4 = *reinterpret_cast<const float4*>(er + b * M_ + sub * 4);
  const float4 a4 = *reinterpret_cast<const float4*>(ad + b * M_ + sub * 4);

  const float* memb = mem + (size_t)b * S_ * M_;
  float* outb       = newmem + (size_t)b * S_ * M_;

  float rx = 0.0f, ry = 0.0f, rz = 0.0f, rw = 0.0f;

  for (int it = 0; it < 32; ++it) {
    int s = chunk * 512 + it * 16 + wave * 2 + half;
    const float4 v = *reinterpret_cast<const float4*>(memb + (size_t)s * M_ + sub * 4);
    float wR = wgr[(size_t)b * S_ + s] * invR;
    float wW = wgw[(size_t)b * S_ + s] * invW;
    rx += wR * v.x; ry += wR * v.y; rz += wR * v.z; rw += wR * v.w;
    float4 o;
    o.x = v.x * (1.0f - wW * e4.x) + wW * a4.x;
    o.y = v.y * (1.0f - wW * e4.y) + wW * a4.y;
    o.z = v.z * (1.0f - wW * e4.z) + wW * a4.z;
    o.w = v.w * (1.0f - wW * e4.w) + wW * a4.w;
    *reinterpret_cast<float4*>(outb + (size_t)s * M_ + sub * 4) = o;
  }

  // fold the two half-wave rows together (same column chunk)
  rx += __shfl_down(rx, 16, 32);
  ry += __shfl_down(ry, 16, 32);
  rz += __shfl_down(rz, 16, 32);
  rw += __shfl_down(rw, 16, 32);

  __shared__ float lds[8][64];
  if (half == 0) {
    lds[wave][sub * 4 + 0] = rx;
    lds[wave][sub * 4 + 1] = ry;
    lds[wave][sub * 4 + 2] = rz;
    lds[wave][sub * 4 + 3] = rw;
  }
  __syncthreads();
  if (threadIdx.x < 64) {
    float s = 0.0f;
    for (int w = 0; w < 8; ++w) s += lds[w][threadIdx.x];
    rpart[((size_t)b * 32 + chunk) * M_ + threadIdx.x] = s;
  }
}

// ---------------------------------------------------------------------------
// K11: r[b,m] = sum over 32 chunk partials (fixed order)
// ---------------------------------------------------------------------------
__global__ void rreduce_kernel(const float* __restrict__ rpart,
                               float* __restrict__ r) {
  int b = blockIdx.x, m = threadIdx.x;
  float s = 0.0f;
  for (int c = 0; c < 32; ++c) s += rpart[((size_t)b * 32 + c) * M_ + m];
  r[b * M_ + m] = s;
}

// ---------------------------------------------------------------------------
// K12: output = softmax(concat(h, r) @ W_o.T + b_o)   (B x 64)
// ---------------------------------------------------------------------------
__global__ void outhead_kernel(const float* __restrict__ h,
                               const float* __restrict__ r,
                               const float* __restrict__ Wo,
                               const float* __restrict__ bo,
                               float* __restrict__ out) {
  int b = blockIdx.x, j = threadIdx.x;   // 64 threads
  const float* wrow = Wo + (size_t)j * EXT_;
  const float* hb = h + (size_t)b * HID_;
  const float* rb = r + (size_t)b * M_;
  float acc = bo[j];
  for (int k = 0; k < HID_; ++k) acc += hb[k] * wrow[k];
  for (int m = 0; m < M_; ++m)  acc += rb[m] * wrow[HID_ + m];

  __shared__ float buf[64];
  buf[j] = acc;
  __syncthreads();
  float mx = -INFINITY;
  for (int k = 0; k < 64; ++k) mx = fmaxf(mx, buf[k]);
  float e = expf(acc - mx);
  __syncthreads();
  buf[j] = e;
  __syncthreads();
  float den = 0.0f;
  for (int k = 0; k < 64; ++k) den += buf[k];
  out[b * 64 + j] = e / den;
}

}  // namespace

// ---------------------------------------------------------------------------
extern "C" void kernel_launch(void* const* d_in, const int* in_sizes, int n_in,
                              void* d_out, int out_size, void* d_ws, size_t ws_size,
                              hipStream_t stream) {
  const float* x_in  = (const float*)d_in[0];
  const float* pdo   = (const float*)d_in[1];
  const float* prv   = (const float*)d_in[2];
  const float* mem   = (const float*)d_in[3];
  const float* W_ih  = (const float*)d_in[4];
  // d_in[5] = W_hh (unused in the reference forward pass)
  const float* b_ih  = (const float*)d_in[6];
  const float* b_hh  = (const float*)d_in[7];
  const float* W_r   = (const float*)d_in[8];
  const float* b_r   = (const float*)d_in[9];
  const float* W_w   = (const float*)d_in[10];
  const float* b_w   = (const float*)d_in[11];
  const float* W_o   = (const float*)d_in[12];
  const float* b_o   = (const float*)d_in[13];

  float* out    = (float*)d_out;            // output (32,64) first
  float* newmem = out + B_ * M_;            // then new_mem (32,16384,64)

  float* ws = (float*)d_ws;
  size_t off = 0;
  auto alloc = [&](size_t n) { float* p = ws + off; off += n; return p; };
  float* xt    = alloc((size_t)B_ * KL_);
  float* z     = alloc((size_t)B_ * NZ_);
  float* h     = alloc((size_t)B_ * HID_);
  float* outr  = alloc((size_t)B_ * NRP_);
  float* outw  = alloc((size_t)B_ * NWP_);
  float* kr    = alloc((size_t)B_ * M_);
  float* kw    = alloc((size_t)B_ * M_);
  float* er    = alloc((size_t)B_ * M_);
  float* ad    = alloc((size_t)B_ * M_);
  float* prm   = alloc((size_t)2 * B_ * 8);
  float* zmax  = alloc(2 * B_);
  float* esum  = alloc(2 * B_);
  float* wsum  = alloc(2 * B_);
  float* zr    = alloc((size_t)B_ * S_);
  float* zw    = alloc((size_t)B_ * S_);
  float* wgr   = alloc((size_t)B_ * S_);
  float* wgw   = alloc((size_t)B_ * S_);
  float* rpart = alloc((size_t)B_ * 32 * M_);
  float* rvec  = alloc((size_t)B_ * M_);

  // 1: concat inputs
  build_xt_kernel<<<(B_ * KL_ + 255) / 256, 256, 0, stream>>>(x_in, pdo, prv, xt);
  // 2: z = xt @ W_ih^T + b_ih + b_hh   (WMMA, 2x128 full tiles, K=192)
  wmma_gemm_nt<<<dim3(NZ_ / 16, 2), 32, 0, stream>>>(xt, W_ih, b_ih, b_hh, z, KL_, NZ_, NZ_);
  // 3: LSTM activations
  lstm_act_kernel<<<(B_ * HID_ + 255) / 256, 256, 0, stream>>>(z, h);
  // 4: projections (WMMA, K=512; only last tile per row is masked)
  wmma_gemm_nt<<<dim3(NRP_ / 16, 2), 32, 0, stream>>>(h, W_r, b_r, nullptr, outr, HID_, NR_, NRP_);
  wmma_gemm_nt<<<dim3(NWP_ / 16, 2), 32, 0, stream>>>(h, W_w, b_w, nullptr, outw, HID_, NW_, NWP_);
  // 5: addressing parameters
  params_kernel<<<1, 32, 0, stream>>>(outr, outw, kr, kw, er, ad, prm);
  // 6: streaming pass 1 (cosine scores, 128 MB read -> lands in L2)
  scores_kernel<<<B_ * 32, 256, 0, stream>>>(mem, kr, kw, prm, zr, zw);
  // 7-9: softmax over S, shift, sharpen
  maxred_kernel<<<2 * B_, RT_, 0, stream>>>(zr, zw, zmax);
  expsum_kernel<<<2 * B_, RT_, 0, stream>>>(zr, zw, zmax, esum);
  shiftpow_kernel<<<2 * B_, RT_, 0, stream>>>(zr, zw, prm, esum, wgr, wgw, wsum);
  // 10: streaming pass 2 (new_mem write + r partials; memory re-read from L2)
  finalize_kernel<<<B_ * 32, 256, 0, stream>>>(mem, wgr, wgw, wsum, er, ad, newmem, rpart);
  // 11: r reduce
  rreduce_kernel<<<B_, 64, 0, stream>>>(rpart, rvec);
  // 12: output head + softmax
  outhead_kernel<<<B_, 64, 0, stream>>>(h, rvec, W_o, b_o, out);

  (void)in_sizes; (void)n_in; (void)out_size; (void)ws_size;
}